// AgnosticStochasticActor_23141283791524
// MI455X (gfx1250) — compile-verified
//
#include <hip/hip_runtime.h>
#include <hip/hip_bf16.h>
#include <cstdint>
#include <cstddef>

// ---------------------------------------------------------------------------
// Model dims (fixed by the reference)
// ---------------------------------------------------------------------------
#define KB      32      // batch
#define KL      256     // encoder sequence length (obs dim)
#define KDM     256     // d_model
#define KDI     512     // d_inner = 2*d_model
#define KDS     16      // d_state
#define KDTR    16      // dt_rank
#define KXP     48      // dt_rank + 2*d_state
#define KACT    64      // act_dim
#define KNB     2       // blocks per stack

typedef float v2f __attribute__((ext_vector_type(2)));
typedef float v8f __attribute__((ext_vector_type(8)));

__device__ __forceinline__ float silu_f(float x) {
    return x / (1.0f + expf(-x));
}
__device__ __forceinline__ float softplus_f(float x) {
    return (x > 20.0f) ? x : log1pf(expf(x));
}

// ---------------------------------------------------------------------------
// WMMA f32 GEMM:  C[M,N] = X[M,K] @ W[N,K]^T   (all row-major)
// One wave -> a 64x16 strip of C: 4 M-tiles share one B (weight) fragment per
// k-step -> 4 independent accumulator chains + 4x weight reuse. The k-loop is
// explicitly software-pipelined (next k-step's 5 fragments are loaded before
// this k-step's 4 WMMAs) and __launch_bounds__(128,1) relaxes the VGPR budget
// so the loads stay double-buffered instead of serializing on one reg pair.
// EPI: 0 = none, 1 = relu, 2 = softplus(x + bias[col])
// Requires: M % 64 == 0, N % 16 == 0, K % 4 == 0 (holds for every GEMM here).
// ---------------------------------------------------------------------------
template <int EPI>
__global__ __launch_bounds__(128, 1)
void gemm_xwt_wmma(const float* __restrict__ X, int lda,
                   const float* __restrict__ W, int ldb,
                   float* __restrict__ C, int ldc,
                   int M, int N, int K,
                   const float* __restrict__ bias)
{
    const int lane = threadIdx.x & 31;
    const int wave = threadIdx.x >> 5;              // 4 waves / block along N
    const int m0 = blockIdx.x << 6;                 // 64-row strip
    const int n0 = (blockIdx.y * 4 + wave) << 4;
    if (n0 >= N || m0 >= M) return;                 // wave-uniform: whole wave exits

    const int l15 = lane & 15;
    const int hi  = lane >> 4;                      // 0: K pair {0,1}, 1: K pair {2,3}
    const int kh  = hi << 1;

    // A fragments: lane = M row within tile, half-wave selects K pair
    const float* xr0 = X + (size_t)(m0 + l15) * lda + kh;
    const float* xr1 = xr0 + (size_t)16 * lda;
    const float* xr2 = xr0 + (size_t)32 * lda;
    const float* xr3 = xr0 + (size_t)48 * lda;
    // B fragment: lane = N col, half-wave selects K pair (W row-major = B^T)
    const float* wr  = W + (size_t)(n0 + l15) * ldb + kh;

    v8f acc0 = {0.f,0.f,0.f,0.f,0.f,0.f,0.f,0.f};
    v8f acc1 = acc0, acc2 = acc0, acc3 = acc0;

    // ---- software pipeline: prologue loads k-step 0 ----
    v2f b_c  = *(const v2f*)(wr);
    v2f a0_c = *(const v2f*)(xr0);
    v2f a1_c = *(const v2f*)(xr1);
    v2f a2_c = *(const v2f*)(xr2);
    v2f a3_c = *(const v2f*)(xr3);

#pragma unroll 4
    for (int k = 4; k < K; k += 4) {
        // prefetch next k-step's fragments (independent of this step's WMMAs)
        v2f b_n  = *(const v2f*)(wr  + k);
        v2f a0_n = *(const v2f*)(xr0 + k);
        v2f a1_n = *(const v2f*)(xr1 + k);
        v2f a2_n = *(const v2f*)(xr2 + k);
        v2f a3_n = *(const v2f*)(xr3 + k);

        acc0 = __builtin_amdgcn_wmma_f32_16x16x4_f32(false, a0_c, false, b_c, (short)0, acc0, false, false);
        acc1 = __builtin_amdgcn_wmma_f32_16x16x4_f32(false, a1_c, false, b_c, (short)0, acc1, false, false);
        acc2 = __builtin_amdgcn_wmma_f32_16x16x4_f32(false, a2_c, false, b_c, (short)0, acc2, false, false);
        acc3 = __builtin_amdgcn_wmma_f32_16x16x4_f32(false, a3_c, false, b_c, (short)0, acc3, false, false);

        b_c = b_n; a0_c = a0_n; a1_c = a1_n; a2_c = a2_n; a3_c = a3_n;
    }
    // ---- epilogue: last k-step ----
    acc0 = __builtin_amdgcn_wmma_f32_16x16x4_f32(false, a0_c, false, b_c, (short)0, acc0, false, false);
    acc1 = __builtin_amdgcn_wmma_f32_16x16x4_f32(false, a1_c, false, b_c, (short)0, acc1, false, false);
    acc2 = __builtin_amdgcn_wmma_f32_16x16x4_f32(false, a2_c, false, b_c, (short)0, acc2, false, false);
    acc3 = __builtin_amdgcn_wmma_f32_16x16x4_f32(false, a3_c, false, b_c, (short)0, acc3, false, false);

    // C/D layout: VGPR j -> M = tile_base + j (+8 for lanes 16..31), N = n0 + (lane&15)
    const int ccol = n0 + l15;
    const float bv = (EPI == 2) ? bias[ccol] : 0.0f;
    v8f accs[4] = {acc0, acc1, acc2, acc3};
#pragma unroll
    for (int t = 0; t < 4; ++t) {
        const int rbase = m0 + (t << 4) + (hi << 3);
#pragma unroll
        for (int j = 0; j < 8; ++j) {
            float v = accs[t][j];
            if (EPI == 1) v = fmaxf(v, 0.0f);
            if (EPI == 2) v = softplus_f(v + bv);
            C[(size_t)(rbase + j) * ldc + ccol] = v;
        }
    }
}

// ---------------------------------------------------------------------------
// Scalar embed: u[tok, dm] = x[tok] * emb_w[dm] + emb_b[dm]
// ---------------------------------------------------------------------------
__global__ void embed_kernel(const float* __restrict__ x,
                             const float* __restrict__ ew,
                             const float* __restrict__ eb,
                             float* __restrict__ u, int total)
{
    int idx = blockIdx.x * blockDim.x + threadIdx.x;
    if (idx >= total) return;
    int dm  = idx & (KDM - 1);
    int tok = idx >> 8;
    u[idx] = x[tok] * ew[dm] + eb[dm];
}

// ---------------------------------------------------------------------------
// Causal depthwise conv (width 4) over time + conv_b + SiLU.
// xz rows are [x(512) | z(512)]; reads the x half, writes xc[tok, d].
// ---------------------------------------------------------------------------
__global__ void conv_silu_kernel(const float* __restrict__ xz,
                                 const float* __restrict__ cw,  // [512,4]
                                 const float* __restrict__ cb,  // [512]
                                 float* __restrict__ xc,
                                 int Lseq, int total)
{
    int idx = blockIdx.x * blockDim.x + threadIdx.x;
    if (idx >= total) return;
    int d   = idx & (KDI - 1);
    int tok = idx >> 9;
    int t   = tok % Lseq;
    const float* w = cw + d * 4;
    float s = cb[d];
#pragma unroll
    for (int j = 0; j < 4; ++j) {
        int tt = t - 3 + j;
        if (tt >= 0)
            s += xz[(size_t)(tok - 3 + j) * (2 * KDI) + d] * w[j];
    }
    xc[idx] = silu_f(s);
}

// ---------------------------------------------------------------------------
// Selective scan. grid = B, block = 512 (one thread per inner channel).
// Each thread carries h[16] in registers across the L sequential steps.
// B_t/C_t (shared across all 512 threads of a block) come from xdbl[tok,16..47]
// and are served by L2 broadcast. Fuses  y = (scan + x*D) * silu(z).
// ---------------------------------------------------------------------------
__global__ __launch_bounds__(512)
void scan_kernel(const float* __restrict__ xc,
                 const float* __restrict__ xz,
                 const float* __restrict__ xdbl,
                 const float* __restrict__ delta,
                 const float* __restrict__ A_log,  // [512,16]
                 const float* __restrict__ Dp,     // [512]
                 float* __restrict__ y,
                 int Lseq)
{
    const int b = blockIdx.x;
    const int d = threadIdx.x;

    float A[KDS];
#pragma unroll
    for (int n = 0; n < KDS; ++n) A[n] = -expf(A_log[d * KDS + n]);
    const float Dv = Dp[d];

    float h[KDS];
#pragma unroll
    for (int n = 0; n < KDS; ++n) h[n] = 0.0f;

    size_t tok = (size_t)b * Lseq;
    for (int t = 0; t < Lseq; ++t, ++tok) {
        float dt = delta[tok * KDI + d];
        float xv = xc[tok * KDI + d];
        const float* bc = xdbl + tok * KXP;
        float dx = dt * xv;
        float acc = 0.0f;
#pragma unroll
        for (int n = 0; n < KDS; ++n) {
            float dA = expf(dt * A[n]);
            h[n] = dA * h[n] + dx * bc[KDTR + n];          // B_t
            acc = fmaf(h[n], bc[KDTR + KDS + n], acc);     // C_t
        }
        float zv = xz[tok * (2 * KDI) + KDI + d];
        y[tok * KDI + d] = (acc + xv * Dv) * silu_f(zv);
    }
}

// ---------------------------------------------------------------------------
// h[b,dm] = relu(mean_l u[b,l,dm]); broadcast into hd[b, 0..ACT-1, dm]
// ---------------------------------------------------------------------------
__global__ void pool_broadcast_kernel(const float* __restrict__ u,
                                      float* __restrict__ hd,
                                      int Lseq, int total)
{
    int idx = blockIdx.x * blockDim.x + threadIdx.x;  // over B*KDM
    if (idx >= total) return;
    int dm = idx & (KDM - 1);
    int b  = idx >> 8;
    float s = 0.0f;
    for (int l = 0; l < Lseq; ++l)
        s += u[(size_t)(b * Lseq + l) * KDM + dm];
    float v = fmaxf(s / (float)Lseq, 0.0f);
    for (int a = 0; a < KACT; ++a)
        hd[(size_t)(b * KACT + a) * KDM + dm] = v;
}

// ---------------------------------------------------------------------------
// out[0,b,a] = mean_dm a_mean;  out[1,b,a] = softplus(mean_dm a_logstd)
// ---------------------------------------------------------------------------
__global__ void final_kernel(const float* __restrict__ am,
                             const float* __restrict__ al,
                             float* __restrict__ out, int total)
{
    int idx = blockIdx.x * blockDim.x + threadIdx.x;  // over B*ACT
    if (idx >= total) return;
    const float* r1 = am + (size_t)idx * KDM;
    const float* r2 = al + (size_t)idx * KDM;
    float s1 = 0.0f, s2 = 0.0f;
    for (int dm = 0; dm < KDM; ++dm) { s1 += r1[dm]; s2 += r2[dm]; }
    out[idx]         = s1 / (float)KDM;
    out[total + idx] = softplus_f(s2 / (float)KDM);
}

// ---------------------------------------------------------------------------
// Host side
// ---------------------------------------------------------------------------
struct StackP {
    const float *in_w, *conv_w, *conv_b, *xproj_w, *dt_w, *dt_b, *A_log, *D, *out_w;
};

static inline StackP make_stackp(void* const* d_in, int base) {
    StackP p;
    p.in_w    = (const float*)d_in[base + 0];
    p.conv_w  = (const float*)d_in[base + 1];
    p.conv_b  = (const float*)d_in[base + 2];
    p.xproj_w = (const float*)d_in[base + 3];
    p.dt_w    = (const float*)d_in[base + 4];
    p.dt_b    = (const float*)d_in[base + 5];
    p.A_log   = (const float*)d_in[base + 6];
    p.D       = (const float*)d_in[base + 7];
    p.out_w   = (const float*)d_in[base + 8];
    return p;
}

static void run_block(hipStream_t st, const StackP& p, int blk,
                      const float* uin, float* uout,
                      int Bn, int Lseq, bool relu_out,
                      float* xz, float* xc, float* xdbl, float* delta, float* ybuf)
{
    const int M = Bn * Lseq;                         // multiple of 64
    const float* in_w    = p.in_w    + (size_t)blk * (2 * KDI) * KDM;
    const float* conv_w  = p.conv_w  + (size_t)blk * KDI * 4;
    const float* conv_b  = p.conv_b  + (size_t)blk * KDI;
    const float* xproj_w = p.xproj_w + (size_t)blk * KXP * KDI;
    const float* dt_w    = p.dt_w    + (size_t)blk * KDI * KDTR;
    const float* dt_b    = p.dt_b    + (size_t)blk * KDI;
    const float* A_log   = p.A_log   + (size_t)blk * KDI * KDS;
    const float* Dv      = p.D       + (size_t)blk * KDI;
    const float* out_w   = p.out_w   + (size_t)blk * KDM * KDI;

    // xz = u @ in_w^T           [M,256] x [1024,256]^T -> [M,1024]
    gemm_xwt_wmma<0><<<dim3(M / 64, (2 * KDI) / 64), 128, 0, st>>>(
        uin, KDM, in_w, KDM, xz, 2 * KDI, M, 2 * KDI, KDM, nullptr);

    // xc = silu(causal_conv(x) + conv_b)
    conv_silu_kernel<<<(M * KDI + 255) / 256, 256, 0, st>>>(
        xz, conv_w, conv_b, xc, Lseq, M * KDI);

    // xdbl = xc @ xproj_w^T     [M,512] x [48,512]^T -> [M,48]  (3 N-tiles)
    gemm_xwt_wmma<0><<<dim3(M / 64, 1), 128, 0, st>>>(
        xc, KDI, xproj_w, KDI, xdbl, KXP, M, KXP, KDI, nullptr);

    // delta = softplus(dt @ dt_w^T + dt_b)   dt = xdbl[:, 0:16] (lda = 48)
    gemm_xwt_wmma<2><<<dim3(M / 64, KDI / 64), 128, 0, st>>>(
        xdbl, KXP, dt_w, KDTR, delta, KDI, M, KDI, KDTR, dt_b);

    // selective scan + x*D + silu(z) gate
    scan_kernel<<<Bn, KDI, 0, st>>>(xc, xz, xdbl, delta, A_log, Dv, ybuf, Lseq);

    // uout = y @ out_w^T  (+ optional relu)
    if (relu_out)
        gemm_xwt_wmma<1><<<dim3(M / 64, KDM / 64), 128, 0, st>>>(
            ybuf, KDI, out_w, KDI, uout, KDM, M, KDM, KDI, nullptr);
    else
        gemm_xwt_wmma<0><<<dim3(M / 64, KDM / 64), 128, 0, st>>>(
            ybuf, KDI, out_w, KDI, uout, KDM, M, KDM, KDI, nullptr);
}

extern "C" void kernel_launch(void* const* d_in, const int* in_sizes, int n_in,
                              void* d_out, int out_size, void* d_ws, size_t ws_size,
                              hipStream_t stream)
{
    (void)in_sizes; (void)n_in; (void)out_size; (void)ws_size;

    const float* x     = (const float*)d_in[0];
    const float* emb_w = (const float*)d_in[1];
    const float* emb_b = (const float*)d_in[2];
    StackP enc  = make_stackp(d_in, 3);
    StackP mean = make_stackp(d_in, 12);
    StackP lstd = make_stackp(d_in, 21);
    float* out = (float*)d_out;

    // ---- workspace carve-up (floats) ----
    float* w = (float*)d_ws;
    const size_t nTokEnc = (size_t)KB * KL;          // 8192
    const size_t nTokAct = (size_t)KB * KACT;        // 2048
    float* u0    = w;                      w += nTokEnc * KDM;          // 2M
    float* u1    = w;                      w += nTokEnc * KDM;          // 2M
    float* xz    = w;                      w += nTokEnc * (2 * KDI);    // 8M
    float* xc    = w;                      w += nTokEnc * KDI;          // 4M
    float* xdbl  = w;                      w += nTokEnc * KXP;          // 0.4M
    float* delta = w;                      w += nTokEnc * KDI;          // 4M
    float* ybuf  = w;                      w += nTokEnc * KDI;          // 4M
    float* hd    = w;                      w += nTokAct * KDM;          // 0.5M
    float* t0    = w;                      w += nTokAct * KDM;
    float* t1    = w;                      w += nTokAct * KDM;
    float* t2    = w;                      w += nTokAct * KDM;
    float* t3    = w;                      w += nTokAct * KDM;

    // ---- 1. embed: [B,L] -> [B,L,256] ----
    {
        int total = (int)(nTokEnc * KDM);
        embed_kernel<<<(total + 255) / 256, 256, 0, stream>>>(x, emb_w, emb_b, u0, total);
    }

    // ---- 2. encoder stack (relu after both blocks) ----
    run_block(stream, enc, 0, u0, u1, KB, KL, /*relu=*/true, xz, xc, xdbl, delta, ybuf);
    run_block(stream, enc, 1, u1, u0, KB, KL, /*relu=*/true, xz, xc, xdbl, delta, ybuf);

    // ---- 3. pool over L, relu, broadcast to [B,ACT,256] ----
    pool_broadcast_kernel<<<(KB * KDM + 255) / 256, 256, 0, stream>>>(
        u0, hd, KL, KB * KDM);

    // ---- 4. mean head (relu after first block only) ----
    run_block(stream, mean, 0, hd, t0, KB, KACT, /*relu=*/true,  xz, xc, xdbl, delta, ybuf);
    run_block(stream, mean, 1, t0, t1, KB, KACT, /*relu=*/false, xz, xc, xdbl, delta, ybuf);

    // ---- 5. logstd head ----
    run_block(stream, lstd, 0, hd, t2, KB, KACT, /*relu=*/true,  xz, xc, xdbl, delta, ybuf);
    run_block(stream, lstd, 1, t2, t3, KB, KACT, /*relu=*/false, xz, xc, xdbl, delta, ybuf);

    // ---- 6. reduce to [2,B,ACT] ----
    final_kernel<<<(KB * KACT + 255) / 256, 256, 0, stream>>>(
        t1, t3, out, KB * KACT);
}